// EntityBoundaryPredictor_13030930776499
// MI455X (gfx1250) — compile-verified
//
#include <hip/hip_runtime.h>
#include <hip/hip_bf16.h>
#include <math.h>
#include <stdint.h>

typedef float v2f __attribute__((ext_vector_type(2)));
typedef float v8f __attribute__((ext_vector_type(8)));
typedef unsigned int v4u __attribute__((ext_vector_type(4)));
typedef int v4i __attribute__((ext_vector_type(4)));
typedef int v8i __attribute__((ext_vector_type(8)));

#define DIM_B 4
#define DIM_E 16
#define DIM_T 1024
#define DIM_H 1024
#define KC 64                    // K-chunk staged per TDM transfer
#define NEG_FILL (-10000.0f)

// ---------------------------------------------------------------------------
// TDM: issue a 2-D tile load (fp32 elements) from global memory into LDS.
// Descriptor layout per CDNA5 ISA ch.8:
//   group0: [1:0]=count=1, [63:32]=lds_addr, [120:64]=global tile addr,
//           [127:126]=type=2
//   group1: [17:16]=data_size(2 -> 4B), [79:48]=tensor_dim0,
//           [111:80]=tensor_dim1, [127:112]=tile_dim0, [143:128]=tile_dim1,
//           [207:160]=tensor_dim0_stride ; workgroup_mask=0 (not in cluster)
//   groups 2/3: zero (tile_dim2=0 -> 2-D)
// This toolchain exposes the 6-arg builtin:
//   (uint32x4 g0, int32x8 g1, int32x4 g2, int32x4 g3, int32x8 g4, i32 cpol)
// ---------------------------------------------------------------------------
__device__ __forceinline__ void tdm_load_tile_f32(
    const float* __restrict__ gptr,   // global address of tile start
    unsigned lds_byte_addr,           // LDS byte offset of destination
    unsigned tile_w,                  // contiguous elements per row
    unsigned tile_h,                  // rows
    unsigned tensor_w,                // tensor dim0 (elements)
    unsigned tensor_h,                // tensor dim1 (rows)
    unsigned row_stride_elems) {      // elements between consecutive rows
  const unsigned long long ga = (unsigned long long)(uintptr_t)gptr;
  v4u g0;
  g0[0] = 1u;                                        // count=1, user mode
  g0[1] = lds_byte_addr;
  g0[2] = (unsigned)ga;                              // global_addr[31:0]
  g0[3] = (unsigned)((ga >> 32) & 0x01ffffffu)       // global_addr[56:32]
          | (2u << 30);                              // type=2 ("image")
  v8i g1;
  g1[0] = (int)(2u << 16);                           // data_size = 4 bytes
  g1[1] = (int)((tensor_w & 0xffffu) << 16);         // tensor_dim0[15:0]
  g1[2] = (int)(((tensor_w >> 16) & 0xffffu)         // tensor_dim0[31:16]
          | ((tensor_h & 0xffffu) << 16));           // tensor_dim1[15:0]
  g1[3] = (int)(((tensor_h >> 16) & 0xffffu)         // tensor_dim1[31:16]
          | ((tile_w & 0xffffu) << 16));             // tile_dim0
  g1[4] = (int)(tile_h & 0xffffu);                   // tile_dim1 (tile_dim2=0)
  g1[5] = (int)row_stride_elems;                     // dim0_stride[31:0]
  g1[6] = 0;                                         // dim0_stride[47:32] etc.
  g1[7] = 0;
  const v4i z4 = (v4i)0;
  const v8i z8 = (v8i)0;
  __builtin_amdgcn_tensor_load_to_lds(g0, g1, z4, z4, z8, /*cpol=*/0);
}

__device__ __forceinline__ unsigned lds_addr_of(const void* p) {
  // LDS aperture: generic addr[31:0] == LDS byte offset (ISA 10.2)
  return (unsigned)(uintptr_t)p;
}

// ---------------------------------------------------------------------------
// Kernel A: Out[M,N] = A[M,K] @ W[K,N] + bias[N], fp32 WMMA, TDM-staged LDS
// double buffering. Block = 256 threads = 8 waves; block tile = 32 x 64;
// wave (rowT = wave&1, colT = wave>>1) owns one 16x16 WMMA accumulator.
// Wave 0 drives the TDM pipeline: transfer of chunk c+1 overlaps the 16-WMMA
// burst on chunk c; S_WAIT_TENSORCNT 0 + workgroup barrier publish the tile.
// ---------------------------------------------------------------------------
__global__ void gemm_bias_wmma_f32(const float* __restrict__ A,
                                   const float* __restrict__ W,
                                   const float* __restrict__ bias,
                                   float* __restrict__ Out,
                                   int M, int N, int K) {
  __shared__ float sA[2][32 * KC];   // 32 rows x KC   (16 KB)
  __shared__ float sW[2][KC * 64];   // KC rows x 64   (32 KB)

  const int lane = threadIdx.x & 31;
  const int wave = threadIdx.x >> 5;
  const int m    = lane & 15;
  const int kh   = lane >> 4;               // A: K{0,1} vs K{2,3} lane half
  const int rowT = wave & 1;
  const int colT = wave >> 1;
  const int rowBase = blockIdx.y << 5;
  const int colBase = blockIdx.x << 6;
  const int row0 = rowBase + (rowT << 4);
  const int col0 = colBase + (colT << 4);
  const int NC = K / KC;
  const bool leader = (wave == 0);

  if (leader) {
    tdm_load_tile_f32(A + (size_t)rowBase * K, lds_addr_of(&sA[0][0]),
                      KC, 32, K, M, K);
    tdm_load_tile_f32(W + colBase, lds_addr_of(&sW[0][0]),
                      64, KC, N, K, N);
  }

  v8f acc = (v8f)0.0f;
  for (int c = 0; c < NC; ++c) {
    if (leader) __builtin_amdgcn_s_wait_tensorcnt(0);
    __syncthreads();                       // chunk c resident in buf c&1

    if (leader && (c + 1 < NC)) {          // prefetch chunk c+1 into buf ^1
      const int k  = (c + 1) * KC;
      const int nb = (c + 1) & 1;
      tdm_load_tile_f32(A + (size_t)rowBase * K + k, lds_addr_of(&sA[nb][0]),
                        KC, 32, K, M, K);
      tdm_load_tile_f32(W + (size_t)k * N + colBase, lds_addr_of(&sW[nb][0]),
                        64, KC, N, K, N);
    }

    const float* a  = sA[c & 1] + ((rowT << 4) + m) * KC + (kh << 1);
    const float* ww = sW[c & 1] + (colT << 4) + m;
#pragma unroll
    for (int kk = 0; kk < KC; kk += 4) {
      v2f av, bv;
      av.x = a[kk];                        // A[row][kk + 2*kh + 0]
      av.y = a[kk + 1];                    // A[row][kk + 2*kh + 1]
      bv.x = ww[(kk + kh) * 64];           // W[kk + kh    ][col]
      bv.y = ww[(kk + 2 + kh) * 64];       // W[kk + kh + 2][col]
      acc = __builtin_amdgcn_wmma_f32_16x16x4_f32(
          false, av, false, bv, (short)0, acc, false, false);
    }
    __syncthreads();                       // done reading buf c&1
  }

#pragma unroll
  for (int v = 0; v < 8; ++v) {
    const int r = row0 + v + (kh << 3);
    const int c = col0 + m;
    Out[(size_t)r * N + c] = acc[v] + bias[c];
  }
}

// ---------------------------------------------------------------------------
// Kernel B: fused  cls[b,e,t] = sum_h relu(t[b,t,h] + e[b,e,h]) * Wp[h] + bp
//                  cls = mask ? cls : -10000 ;  p = sigmoid(cls)
// One wave per token row; 16 register accumulators (one per entity); e-matrix
// (256 KB) and Wp stay hot in the 192 MB L2, so HBM traffic is one streaming
// read of t. wave32 shuffle reduction at the end.
// ---------------------------------------------------------------------------
__global__ void relu_dot_sigmoid(const float* __restrict__ Tm,   // [B*T, H]
                                 const float* __restrict__ Em,   // [B*E, H]
                                 const float* __restrict__ Wp,   // [H]
                                 const float* __restrict__ bp,   // [1]
                                 const int* __restrict__ mask,   // [B*T]
                                 float* __restrict__ out) {      // cls | p
  const int lane = threadIdx.x & 31;
  const int wave = threadIdx.x >> 5;
  const int b    = blockIdx.y;
  const int t    = (blockIdx.x << 3) + wave;

  const float* trow  = Tm + ((size_t)b * DIM_T + t) * DIM_H;
  const float* ebase = Em + (size_t)b * DIM_E * DIM_H;

  float acc[DIM_E];
#pragma unroll
  for (int e = 0; e < DIM_E; ++e) acc[e] = 0.0f;

  for (int h = lane; h < DIM_H; h += 32) {
    const float tv = trow[h];
    const float wv = Wp[h];
#pragma unroll
    for (int e = 0; e < DIM_E; ++e) {
      float s = tv + ebase[(size_t)e * DIM_H + h];
      s = s > 0.0f ? s : 0.0f;
      acc[e] = fmaf(s, wv, acc[e]);
    }
  }

#pragma unroll
  for (int e = 0; e < DIM_E; ++e) {
    float v = acc[e];
#pragma unroll
    for (int off = 16; off > 0; off >>= 1) v += __shfl_xor(v, off, 32);
    acc[e] = v;
  }

  if (lane < DIM_E) {
    const int e = lane;
    float cls = acc[e] + bp[0];
    if (mask[b * DIM_T + t] == 0) cls = NEG_FILL;
    const float p = 1.0f / (1.0f + __expf(-cls));
    const size_t idx = ((size_t)b * DIM_E + e) * DIM_T + t;
    const size_t BET = (size_t)DIM_B * DIM_E * DIM_T;
    out[idx]       = cls;
    out[BET + idx] = p;
  }
}

extern "C" void kernel_launch(void* const* d_in, const int* in_sizes, int n_in,
                              void* d_out, int out_size, void* d_ws, size_t ws_size,
                              hipStream_t stream) {
  const float* tok  = (const float*)d_in[0];  // [B,T,H]
  const float* ent  = (const float*)d_in[1];  // [B,E,H]
  const int*   mask = (const int*)d_in[2];    // [B,T]
  const float* Wt   = (const float*)d_in[3];  // [H,H]
  const float* bt   = (const float*)d_in[4];  // [H]
  const float* We   = (const float*)d_in[5];  // [H,H]
  const float* be   = (const float*)d_in[6];  // [H]
  const float* Wp   = (const float*)d_in[7];  // [H]
  const float* bp   = (const float*)d_in[8];  // [1]
  float* out = (float*)d_out;

  float* tbuf = (float*)d_ws;                              // [B*T, H] = 16 MB
  float* ebuf = tbuf + (size_t)DIM_B * DIM_T * DIM_H;      // [B*E, H] = 256 KB

  // t = token @ Wt + bt   (M = B*T = 4096; block tile 32x64)
  gemm_bias_wmma_f32<<<dim3(DIM_H / 64, (DIM_B * DIM_T) / 32), dim3(256), 0, stream>>>(
      tok, Wt, bt, tbuf, DIM_B * DIM_T, DIM_H, DIM_H);

  // e = entity @ We + be  (M = B*E = 64)
  gemm_bias_wmma_f32<<<dim3(DIM_H / 64, (DIM_B * DIM_E) / 32), dim3(256), 0, stream>>>(
      ent, We, be, ebuf, DIM_B * DIM_E, DIM_H, DIM_H);

  // fused relu-dot + mask + sigmoid
  relu_dot_sigmoid<<<dim3(DIM_T / 8, DIM_B), dim3(256), 0, stream>>>(
      tbuf, ebuf, Wp, bp, mask, out);
}